// Model_72980084293789
// MI455X (gfx1250) — compile-verified
//
#include <hip/hip_runtime.h>
#include <hip/hip_bf16.h>
#include <math.h>

// ---------------------------------------------------------------------------
// Soft-silhouette rasterization loss for MI455X (gfx1250, wave32).
//
// Roofline: ~51M (pixel,face) pairs * ~45 VALU ops + exp/log each; HBM
// traffic ~260KB. Entirely VALU/transcendental bound -> minimize inner-loop
// ops (precompute per-edge linear coefficients), keep face constants in LDS
// (broadcast reads), use hardware exp/log. The only true matmul (vertex
// transform) runs on V_WMMA_F32_16X16X4_F32.
// ---------------------------------------------------------------------------

#define EPSF      1e-9f
#define INV_SIGMA 1e4f        // 1/SIGMA, SIGMA = 1e-4
#define NV        1024        // vertices
#define NF        1024        // faces
#define IMG       224
#define NPIX      (IMG*IMG)   // 50176 = 196*256
#define FC_STRIDE 28          // floats per face (27 used + 1 pad, 112B = 16B aligned)

typedef __attribute__((ext_vector_type(2))) float v2f;
typedef __attribute__((ext_vector_type(8))) float v8f;

// ---------------------------------------------------------------------------
// Kernel 1: vertex transform + projection via WMMA f32 16x16x4.
// D(16x16) = A(16x4: [x y z 1] per vertex) * B(4x16: cols 0..2 = rows of
// (K*R) with K*t in the K=3 slot).  64 tiles of 16 vertices = 8 blocks * 8
// waves.  D tile bounced through LDS to regroup per-vertex components.
// ---------------------------------------------------------------------------
__global__ void vert_transform_kernel(const float* __restrict__ verts,
                                      const float* __restrict__ quat,
                                      const float* __restrict__ tvec,
                                      const float* __restrict__ Kmat,
                                      float* __restrict__ verts2d) {
    __shared__ float tile[8][256];   // one 16x16 f32 tile per wave
    const int tid  = threadIdx.x;
    const int lane = tid & 31;
    const int wave = tid >> 5;

    // --- quaternion -> rotation (every thread computes; trivial cost) ---
    float qw = quat[0], qx = quat[1], qy = quat[2], qz = quat[3];
    float qn = rsqrtf(qw*qw + qx*qx + qy*qy + qz*qz);
    qw *= qn; qx *= qn; qy *= qn; qz *= qn;
    float R[3][3];
    R[0][0] = 1.f - 2.f*(qy*qy + qz*qz); R[0][1] = 2.f*(qx*qy - qw*qz); R[0][2] = 2.f*(qx*qz + qw*qy);
    R[1][0] = 2.f*(qx*qy + qw*qz); R[1][1] = 1.f - 2.f*(qx*qx + qz*qz); R[1][2] = 2.f*(qy*qz - qw*qx);
    R[2][0] = 2.f*(qx*qz - qw*qy); R[2][1] = 2.f*(qy*qz + qw*qx); R[2][2] = 1.f - 2.f*(qx*qx + qy*qy);

    // M = K*R,  tK = K*t   (proj = M*v + tK)
    float M[3][3], tK[3];
    #pragma unroll
    for (int n = 0; n < 3; ++n) {
        #pragma unroll
        for (int k = 0; k < 3; ++k)
            M[n][k] = Kmat[n*3+0]*R[0][k] + Kmat[n*3+1]*R[1][k] + Kmat[n*3+2]*R[2][k];
        tK[n] = Kmat[n*3+0]*tvec[0] + Kmat[n*3+1]*tvec[1] + Kmat[n*3+2]*tvec[2];
    }

    // --- A operand: lanes 0-15 hold (x,y) K=0,1 for vertex M=lane;
    //               lanes 16-31 hold (z,1) K=2,3 for vertex M=lane-16. ---
    const int  row   = lane & 15;
    const bool lo    = (lane < 16);
    const int  vtile = blockIdx.x * 8 + wave;
    const int  v     = vtile * 16 + row;
    const float vx = verts[3*v], vy = verts[3*v + 1], vz = verts[3*v + 2];
    v2f a;
    a.x = lo ? vx : vz;
    a.y = lo ? vy : 1.0f;

    // --- B operand: lane n<16 holds B[0][n],B[1][n]; lane n+16 holds
    //     B[2][n],B[3][n].  B[k][n] = M[n][k] (n<3), B[3][n] = tK[n]. ---
    const int n = lane & 15;
    float bx0 = lo ? M[0][0] : M[0][2], by0 = lo ? M[0][1] : tK[0];
    float bx1 = lo ? M[1][0] : M[1][2], by1 = lo ? M[1][1] : tK[1];
    float bx2 = lo ? M[2][0] : M[2][2], by2 = lo ? M[2][1] : tK[2];
    v2f b;
    b.x = (n == 0) ? bx0 : (n == 1) ? bx1 : (n == 2) ? bx2 : 0.f;
    b.y = (n == 0) ? by0 : (n == 1) ? by1 : (n == 2) ? by2 : 0.f;

    v8f c = {};
    // 8 args: (neg_a, A, neg_b, B, c_mod, C, reuse_a, reuse_b)
    c = __builtin_amdgcn_wmma_f32_16x16x4_f32(false, a, false, b,
                                              (short)0, c, false, false);

    // D layout: VGPR r -> lanes 0-15: (M=r, N=lane), lanes 16-31: (M=r+8, N=lane-16)
    #pragma unroll
    for (int r = 0; r < 8; ++r) {
        int m = lo ? r : (r + 8);
        tile[wave][m * 16 + (lane & 15)] = c[r];
    }
    __syncthreads();

    // 128 vertices per block: regroup columns 0..2 into (xn, yn)
    if (tid < 128) {
        int wv = tid >> 4, rw = tid & 15;
        int gv = blockIdx.x * 128 + tid;
        float X = tile[wv][rw * 16 + 0];
        float Y = tile[wv][rw * 16 + 1];
        float Z = tile[wv][rw * 16 + 2];
        float inv = 1.f / (Z + EPSF);
        // xn = 2*(X/Z - 112)/224 = (X/Z)/112 - 1
        verts2d[2*gv]     = (X * inv) * (1.f / 112.f) - 1.f;
        verts2d[2*gv + 1] = (Y * inv) * (1.f / 112.f) - 1.f;
    }
}

// ---------------------------------------------------------------------------
// Kernel 2: per-face edge coefficients.  Every per-(pixel,edge) quantity is
// linear in (px, py, rr=px^2+py^2), so hoist 9 constants per edge:
//   cross = abx*py - aby*px + crossC
//   dot   = abx*px + aby*py + dotC
//   pp    = rr + ax2*px + ay2*py + ppC        (ax2=-2ax, ay2=-2ay)
// ---------------------------------------------------------------------------
__global__ void face_const_kernel(const int* __restrict__ faces,
                                  const float* __restrict__ verts2d,
                                  float* __restrict__ fc) {
    int f = blockIdx.x * blockDim.x + threadIdx.x;
    if (f >= NF) return;
    int   i[3] = { faces[3*f], faces[3*f + 1], faces[3*f + 2] };
    float vx[3], vy[3];
    #pragma unroll
    for (int k = 0; k < 3; ++k) {
        vx[k] = verts2d[2*i[k]];
        vy[k] = verts2d[2*i[k] + 1];
    }
    float* o = fc + f * FC_STRIDE;
    #pragma unroll
    for (int k = 0; k < 3; ++k) {
        int   k1 = (k + 1) % 3;
        float Ax = vx[k], Ay = vy[k], Bx = vx[k1], By = vy[k1];
        float abx = Bx - Ax, aby = By - Ay;
        float ab2 = abx*abx + aby*aby;
        o[k*9 + 0] = abx;
        o[k*9 + 1] = aby;
        o[k*9 + 2] = ab2;
        o[k*9 + 3] = 1.f / (ab2 + EPSF);
        o[k*9 + 4] = aby*Ax - abx*Ay;          // crossC
        o[k*9 + 5] = -(Ax*abx + Ay*aby);       // dotC
        o[k*9 + 6] = -2.f * Ax;                // ax2
        o[k*9 + 7] = -2.f * Ay;                // ay2
        o[k*9 + 8] = Ax*Ax + Ay*Ay;            // ppC
    }
    o[27] = 0.f;
}

// ---------------------------------------------------------------------------
// Kernel 3: rasterizer.  1 thread/pixel, 196 blocks * 256.  Faces staged
// through LDS in 4 chunks of 256 (28KB); uniform LDS addresses broadcast to
// all 32 lanes conflict-free.  Stable log_sigmoid via hardware exp/log.
// Deterministic block tree-reduction of the smooth-L1 terms.
// ---------------------------------------------------------------------------
__global__ void raster_kernel(const float* __restrict__ fc,
                              const float* __restrict__ image_ref,
                              float* __restrict__ partial) {
    __shared__ float sf[256 * FC_STRIDE];
    __shared__ float red[256];
    const int tid   = threadIdx.x;
    const int pixel = blockIdx.x * 256 + tid;
    const int pr    = pixel / IMG;
    const int pc    = pixel - pr * IMG;
    const float py  = (2.f * ((float)pr + 0.5f)) / (float)IMG - 1.f;
    const float px  = (2.f * ((float)pc + 0.5f)) / (float)IMG - 1.f;
    const float rr  = px*px + py*py;

    float S = 0.f;
    for (int ch = 0; ch < 4; ++ch) {
        // cooperative stage: thread tid copies face (ch*256+tid), 7x float4
        const float4* src = (const float4*)(fc + (size_t)(ch * 256 + tid) * FC_STRIDE);
        float4*       dst = (float4*)(sf + tid * FC_STRIDE);
        #pragma unroll
        for (int q = 0; q < 7; ++q) dst[q] = src[q];
        __syncthreads();

        for (int f = 0; f < 256; ++f) {
            const float* e = sf + f * FC_STRIDE;
            float mind2 = 0.f, cmn = 0.f, cmx = 0.f;
            #pragma unroll
            for (int k = 0; k < 3; ++k) {
                float abx = e[k*9+0], aby = e[k*9+1], ab2 = e[k*9+2], inv = e[k*9+3];
                float crC = e[k*9+4], dtC = e[k*9+5], ax2 = e[k*9+6], ay2 = e[k*9+7], ppC = e[k*9+8];
                float cross = fmaf(abx, py, fmaf(-aby, px, crC));
                float dot   = fmaf(abx, px, fmaf(aby, py, dtC));
                float pp    = rr + fmaf(ax2, px, fmaf(ay2, py, ppC));
                float tt    = fminf(fmaxf(dot * inv, 0.f), 1.f);     // v_med3
                float d2    = fmaf(tt, fmaf(tt, ab2, -2.f * dot), pp);
                d2 = fmaxf(d2, 0.f);
                if (k == 0) { mind2 = d2; cmn = cross; cmx = cross; }
                else { mind2 = fminf(mind2, d2); cmn = fminf(cmn, cross); cmx = fmaxf(cmx, cross); }
            }
            float sgn = ((cmn >= 0.f) || (cmx <= 0.f)) ? 1.f : -1.f;
            float z   = -sgn * mind2 * INV_SIGMA;
            // log_sigmoid(z) = min(z,0) - log1p(exp(-|z|))
            float ls  = fminf(z, 0.f) - __logf(1.f + __expf(-fabsf(z)));
            S += ls;
        }
        __syncthreads();
    }

    float sil = 1.f - __expf(S);
    float d   = sil - image_ref[pixel];
    float ad  = fabsf(d);
    red[tid]  = (ad < 1.f) ? (0.5f * d * d) : (ad - 0.5f);
    __syncthreads();
    #pragma unroll
    for (int s = 128; s > 0; s >>= 1) {
        if (tid < s) red[tid] += red[tid + s];
        __syncthreads();
    }
    if (tid == 0) partial[blockIdx.x] = red[0];
}

// ---------------------------------------------------------------------------
// Kernel 4: deterministic final reduction of 196 partials, *10, scalar out.
// ---------------------------------------------------------------------------
__global__ void final_reduce_kernel(const float* __restrict__ partial,
                                    float* __restrict__ out, int n) {
    __shared__ float red[256];
    int tid  = threadIdx.x;
    red[tid] = (tid < n) ? partial[tid] : 0.f;
    __syncthreads();
    #pragma unroll
    for (int s = 128; s > 0; s >>= 1) {
        if (tid < s) red[tid] += red[tid + s];
        __syncthreads();
    }
    if (tid == 0) out[0] = 10.f * red[0];
}

// ---------------------------------------------------------------------------
extern "C" void kernel_launch(void* const* d_in, const int* in_sizes, int n_in,
                              void* d_out, int out_size, void* d_ws, size_t ws_size,
                              hipStream_t stream) {
    (void)in_sizes; (void)n_in; (void)out_size; (void)ws_size;
    const float* verts     = (const float*)d_in[0];   // (1,1024,3) f32
    const int*   faces     = (const int*)  d_in[1];   // (1,1024,3) i32
    const float* quat      = (const float*)d_in[2];   // (4,)       f32
    const float* tvec      = (const float*)d_in[3];   // (3,)       f32
    const float* Kmat      = (const float*)d_in[4];   // (3,3)      f32
    const float* image_ref = (const float*)d_in[5];   // (224,224)  f32

    float* ws      = (float*)d_ws;
    float* verts2d = ws;                               // 1024*2      = 2048 f
    float* fc      = ws + 2048;                        // 1024*28     = 28672 f (16B aligned)
    float* partial = ws + 2048 + 28672;                // 196 f

    vert_transform_kernel<<<8,   256, 0, stream>>>(verts, quat, tvec, Kmat, verts2d);
    face_const_kernel    <<<4,   256, 0, stream>>>(faces, verts2d, fc);
    raster_kernel        <<<196, 256, 0, stream>>>(fc, image_ref, partial);
    final_reduce_kernel  <<<1,   256, 0, stream>>>(partial, (float*)d_out, 196);
}